// StructuralAttentionLayer_48395691491596
// MI455X (gfx1250) — compile-verified
//
#include <hip/hip_runtime.h>
#include <hip/hip_bf16.h>
#include <math.h>

// ---------------------------------------------------------------------------
// GAT layer for MI455X (gfx1250, wave32).
//   N=50000 nodes, E=1.6M edges, IN=128, OUT=128, H=8 heads, C=16 head dim.
// GEMMs use V_WMMA_F32_16X16X4_F32 (fp32-exact matrix path on CDNA5).
// Edge phase is L2-resident gather/scatter with float atomics.
// ---------------------------------------------------------------------------

typedef float v2f __attribute__((ext_vector_type(2)));
typedef float v8f __attribute__((ext_vector_type(8)));

#define DIM   128
#define HEADS 8
#define HDIM  16

// ---------------------------------------------------------------------------
// Fused dual GEMM:  x = og_x @ W   and   resid = og_x @ W_res
// One block = 16 rows (M-tile). 4 waves; waves 0-1 compute W's 8 N-tiles,
// waves 2-3 compute W_res's 8 N-tiles (4 tiles of 16x16 per wave).
// A tile (16x128 fp32 = 8KB) staged in LDS once, reused by all waves.
// Store epilogue uses a single uniform bounds check so the fast path is
// branch-free (no per-element exec-mask churn).
// ---------------------------------------------------------------------------
__global__ __launch_bounds__(128) void gat_gemm_wmma(
    const float* __restrict__ og_x,
    const float* __restrict__ W,
    const float* __restrict__ Wres,
    float* __restrict__ x,
    float* __restrict__ resid,
    int n_nodes)
{
    __shared__ float Atile[16 * DIM];       // 8 KB

    const int m0  = blockIdx.x * 16;
    const int tid = threadIdx.x;

    // Cooperative load of 16x128 A tile as float4 (512 float4s / 128 threads).
    float4* At4 = (float4*)Atile;
    for (int i = tid; i < 512; i += 128) {
        int row  = i >> 5;        // 32 float4 per row
        int col4 = i & 31;
        int gr   = m0 + row;
        if (gr >= n_nodes) gr = n_nodes - 1;   // clamp (N%16==0 in practice)
        At4[i] = ((const float4*)(og_x + (size_t)gr * DIM))[col4];
    }
    __syncthreads();

    const int wave = tid >> 5;    // 0..3
    const int lane = tid & 31;
    const int lr   = lane & 15;   // row (A) / col (B,D) within 16-tile
    const int hi   = lane >> 4;   // selects K pair (A/B) and M half (D)

    const float* Bmat  = (wave < 2) ? W : Wres;
    const int    nbase = (wave & 1) * 64;      // cols [0,64) or [64,128)

    v8f acc[4] = {};   // 4 N-tiles of 16x16 f32 accumulators (32 VGPRs)

    for (int k = 0; k < DIM; k += 4) {
        const int kk = k + 2 * hi;
        // A 16x4 fragment: lanes 0-15 hold K={k,k+1}, lanes 16-31 K={k+2,k+3}
        v2f a;
        a.x = Atile[lr * DIM + kk];
        a.y = Atile[lr * DIM + kk + 1];
#pragma unroll
        for (int t = 0; t < 4; ++t) {
            const int n0 = nbase + t * 16;
            // B 4x16 fragment: VGPR0 = row kk, VGPR1 = row kk+1 (per half-wave)
            v2f b;
            b.x = Bmat[(size_t)kk       * DIM + n0 + lr];
            b.y = Bmat[(size_t)(kk + 1) * DIM + n0 + lr];
            acc[t] = __builtin_amdgcn_wmma_f32_16x16x4_f32(
                /*neg_a=*/false, a, /*neg_b=*/false, b,
                /*c_mod=*/(short)0, acc[t],
                /*reuse_a=*/false, /*reuse_b=*/false);
        }
    }

    // D layout: VGPR r -> (M = r + 8*hi, N = lr)
    float* Out = (wave < 2) ? x : resid;
    if (m0 + 16 <= n_nodes) {
        // Fast path: uniform, branch-free stores.
#pragma unroll
        for (int t = 0; t < 4; ++t) {
            const int n0 = nbase + t * 16;
#pragma unroll
            for (int r = 0; r < 8; ++r) {
                const int m = m0 + r + 8 * hi;
                Out[(size_t)m * DIM + n0 + lr] = acc[t][r];
            }
        }
    } else {
        // Tail path (never taken for N % 16 == 0).
#pragma unroll
        for (int t = 0; t < 4; ++t) {
            const int n0 = nbase + t * 16;
#pragma unroll
            for (int r = 0; r < 8; ++r) {
                const int m = m0 + r + 8 * hi;
                if (m < n_nodes)
                    Out[(size_t)m * DIM + n0 + lr] = acc[t][r];
            }
        }
    }
}

// ---------------------------------------------------------------------------
// Init: seg_max = -inf, seg_sum = 0, agg = 0
// ---------------------------------------------------------------------------
__global__ void gat_init(float* __restrict__ seg_max,
                         float* __restrict__ seg_sum,
                         float* __restrict__ agg,
                         int n_nodes)
{
    const int i = blockIdx.x * blockDim.x + threadIdx.x;
    const int nh = n_nodes * HEADS;
    if (i < nh) { seg_max[i] = -INFINITY; seg_sum[i] = 0.0f; }
    if (i < n_nodes * DIM) agg[i] = 0.0f;
}

// ---------------------------------------------------------------------------
// Per-(node,head) attention logits: alpha_l/r[n,h] = dot(x[n,h,:], att_l/r[h])
// ---------------------------------------------------------------------------
__global__ void gat_alpha(const float* __restrict__ x,
                          const float* __restrict__ att_l,
                          const float* __restrict__ att_r,
                          float* __restrict__ alpha_l,
                          float* __restrict__ alpha_r,
                          int n_nodes)
{
    const int idx = blockIdx.x * blockDim.x + threadIdx.x;  // node*8 + head
    if (idx >= n_nodes * HEADS) return;
    const int node = idx >> 3, h = idx & 7;
    const float* xr = x + (size_t)node * DIM + h * HDIM;
    const float* al = att_l + h * HDIM;
    const float* ar = att_r + h * HDIM;
    float sl = 0.0f, sr = 0.0f;
#pragma unroll
    for (int c = 0; c < HDIM; ++c) {
        const float v = xr[c];
        sl = fmaf(v, al[c], sl);
        sr = fmaf(v, ar[c], sr);
    }
    alpha_l[idx] = sl;
    alpha_r[idx] = sr;
}

__device__ __forceinline__ float leaky02(float a) {
    return a > 0.0f ? a : 0.2f * a;
}

// Order-preserving float atomic max (int trick; seg_max initialized to -inf).
__device__ __forceinline__ void atomicMaxF(float* addr, float val) {
    if (val >= 0.0f)
        atomicMax((int*)addr, __float_as_int(val));
    else
        atomicMin((unsigned int*)addr, __float_as_uint(val));
}

// ---------------------------------------------------------------------------
// Edge pass 1: segment max of leaky-relu'd logits (per head)
// ---------------------------------------------------------------------------
__global__ void gat_edge_max(const int* __restrict__ src,
                             const int* __restrict__ dst,
                             const float* __restrict__ ew,
                             const float* __restrict__ alpha_l,
                             const float* __restrict__ alpha_r,
                             float* __restrict__ seg_max,
                             int n_edges)
{
    const int e = blockIdx.x * blockDim.x + threadIdx.x;
    if (e >= n_edges) return;
    const int s = src[e], d = dst[e];
    const float w = ew[e];
    const float* al = alpha_l + (size_t)s * HEADS;
    const float* ar = alpha_r + (size_t)d * HEADS;
    float* sm = seg_max + (size_t)d * HEADS;
#pragma unroll
    for (int h = 0; h < HEADS; ++h) {
        const float a = leaky02(w * (al[h] + ar[h]));
        atomicMaxF(&sm[h], a);
    }
}

// ---------------------------------------------------------------------------
// Edge pass 2: seg_sum[dst,h] += exp(a - seg_max[dst,h])
// ---------------------------------------------------------------------------
__global__ void gat_edge_sum(const int* __restrict__ src,
                             const int* __restrict__ dst,
                             const float* __restrict__ ew,
                             const float* __restrict__ alpha_l,
                             const float* __restrict__ alpha_r,
                             const float* __restrict__ seg_max,
                             float* __restrict__ seg_sum,
                             int n_edges)
{
    const int e = blockIdx.x * blockDim.x + threadIdx.x;
    if (e >= n_edges) return;
    const int s = src[e], d = dst[e];
    const float w = ew[e];
    const float* al = alpha_l + (size_t)s * HEADS;
    const float* ar = alpha_r + (size_t)d * HEADS;
    const float* sm = seg_max + (size_t)d * HEADS;
    float* ss = seg_sum + (size_t)d * HEADS;
#pragma unroll
    for (int h = 0; h < HEADS; ++h) {
        const float a = leaky02(w * (al[h] + ar[h]));
        float m = sm[h];
        if (!isfinite(m)) m = 0.0f;
        atomicAdd(&ss[h], expf(a - m));
    }
}

// ---------------------------------------------------------------------------
// Edge pass 3 (heavy): agg[dst] += coef * x[src].
// 4 threads per edge; each handles 2 heads = 32 columns (float4 loads of x,
// per-float global_atomic_add_f32 into agg — L2-resident).
// ---------------------------------------------------------------------------
__global__ void gat_edge_agg(const int* __restrict__ src,
                             const int* __restrict__ dst,
                             const float* __restrict__ ew,
                             const float* __restrict__ alpha_l,
                             const float* __restrict__ alpha_r,
                             const float* __restrict__ seg_max,
                             const float* __restrict__ seg_sum,
                             const float* __restrict__ x,
                             float* __restrict__ agg,
                             int n_edges)
{
    const long long gid = (long long)blockIdx.x * blockDim.x + threadIdx.x;
    const long long e   = gid >> 2;
    if (e >= n_edges) return;
    const int part = (int)(gid & 3);
    const int s = src[e], d = dst[e];
    const float w = ew[e];
#pragma unroll
    for (int j = 0; j < 2; ++j) {
        const int h = part * 2 + j;
        const float a = leaky02(w * (alpha_l[(size_t)s * HEADS + h] +
                                     alpha_r[(size_t)d * HEADS + h]));
        float m = seg_max[(size_t)d * HEADS + h];
        if (!isfinite(m)) m = 0.0f;
        const float coef = expf(a - m) /
                           (seg_sum[(size_t)d * HEADS + h] + 1e-16f);
        const float4* xs = (const float4*)(x + (size_t)s * DIM + h * HDIM);
        float* ag = agg + (size_t)d * DIM + h * HDIM;
#pragma unroll
        for (int c4 = 0; c4 < 4; ++c4) {
            const float4 v = xs[c4];
            atomicAdd(&ag[c4 * 4 + 0], v.x * coef);
            atomicAdd(&ag[c4 * 4 + 1], v.y * coef);
            atomicAdd(&ag[c4 * 4 + 2], v.z * coef);
            atomicAdd(&ag[c4 * 4 + 3], v.w * coef);
        }
    }
}

// ---------------------------------------------------------------------------
// Finish: out = elu(agg) + resid   (resid already sitting in d_out)
// ---------------------------------------------------------------------------
__global__ void gat_finish(const float* __restrict__ agg,
                           float* __restrict__ out,
                           int total)
{
    const int i = blockIdx.x * blockDim.x + threadIdx.x;
    if (i >= total) return;
    const float a = agg[i];
    const float e = a > 0.0f ? a : (expf(a) - 1.0f);
    out[i] = e + out[i];
}

// ---------------------------------------------------------------------------
extern "C" void kernel_launch(void* const* d_in, const int* in_sizes, int n_in,
                              void* d_out, int out_size, void* d_ws, size_t ws_size,
                              hipStream_t stream)
{
    const float* og_x       = (const float*)d_in[0];
    const int*   edge_index = (const int*)  d_in[1];
    const float* edge_w     = (const float*)d_in[2];
    const float* W          = (const float*)d_in[3];
    const float* att_l      = (const float*)d_in[4];
    const float* att_r      = (const float*)d_in[5];
    const float* W_res      = (const float*)d_in[6];
    float*       out        = (float*)d_out;

    const int n_nodes = in_sizes[0] / DIM;
    const int n_edges = in_sizes[2];
    const int* src = edge_index;
    const int* dst = edge_index + n_edges;

    // Workspace layout (floats): x | agg | alpha_l | alpha_r | seg_max | seg_sum
    float* ws      = (float*)d_ws;
    float* x       = ws;
    float* agg     = x       + (size_t)n_nodes * DIM;
    float* alpha_l = agg     + (size_t)n_nodes * DIM;
    float* alpha_r = alpha_l + (size_t)n_nodes * HEADS;
    float* seg_max = alpha_r + (size_t)n_nodes * HEADS;
    float* seg_sum = seg_max + (size_t)n_nodes * HEADS;

    const int TB = 256;

    // 1) init accumulators
    {
        const int total = n_nodes * DIM;
        gat_init<<<(total + TB - 1) / TB, TB, 0, stream>>>(seg_max, seg_sum, agg, n_nodes);
    }
    // 2) fused dual GEMM via WMMA (x -> ws, residual -> d_out)
    {
        const int mtiles = (n_nodes + 15) / 16;
        gat_gemm_wmma<<<mtiles, 128, 0, stream>>>(og_x, W, W_res, x, out, n_nodes);
    }
    // 3) per-node attention logits
    {
        const int total = n_nodes * HEADS;
        gat_alpha<<<(total + TB - 1) / TB, TB, 0, stream>>>(x, att_l, att_r,
                                                            alpha_l, alpha_r, n_nodes);
    }
    // 4) segment max
    gat_edge_max<<<(n_edges + TB - 1) / TB, TB, 0, stream>>>(src, dst, edge_w,
                                                             alpha_l, alpha_r,
                                                             seg_max, n_edges);
    // 5) segment sum of exp
    gat_edge_sum<<<(n_edges + TB - 1) / TB, TB, 0, stream>>>(src, dst, edge_w,
                                                             alpha_l, alpha_r,
                                                             seg_max, seg_sum, n_edges);
    // 6) weighted message aggregation
    {
        const long long total = (long long)n_edges * 4;
        const long long blocks = (total + TB - 1) / TB;
        gat_edge_agg<<<(unsigned)blocks, TB, 0, stream>>>(src, dst, edge_w,
                                                          alpha_l, alpha_r,
                                                          seg_max, seg_sum,
                                                          x, agg, n_edges);
    }
    // 7) elu + residual
    {
        const int total = n_nodes * DIM;
        gat_finish<<<(total + TB - 1) / TB, TB, 0, stream>>>(agg, out, total);
    }
}